// loss_separation_89867895702076
// MI455X (gfx1250) — compile-verified
//
#include <hip/hip_runtime.h>
#include <math.h>

// CDNA5 / gfx1250: wave32, WMMA (not MFMA).
typedef __attribute__((ext_vector_type(2))) float v2f;
typedef __attribute__((ext_vector_type(8))) float v8f;

#define BATCH       32
#define NKP         8192
#define DIM         128
#define WG_PER_B    32
#define ROWS_PER_WG (NKP / WG_PER_B)   // 256 rows per workgroup
#define THREADS     256                // 8 waves of 32 on gfx1250
#define WAVES       (THREADS / 32)

// Phase 1: each workgroup reduces a [256 x 128] tile of one batch.
//  - Column sums s[b,d] via V_WMMA_F32_16X16X4_F32 with A = ones(16x4):
//    D[m][n] = sum_k B[k][n] + C, i.e. every row of D is the running
//    column-sum vector for this wave's 16 columns.
//  - sum(x^2) via plain VALU fma, then wave/LDS tree reduction.
__global__ __launch_bounds__(THREADS)
void sep_partial_kernel(const float* __restrict__ x,
                        float* __restrict__ ws_s,     // [BATCH*WG_PER_B, DIM]
                        float* __restrict__ ws_ssq)   // [BATCH*WG_PER_B]
{
    const int wg   = blockIdx.x;
    const int b    = wg / WG_PER_B;
    const int g    = wg % WG_PER_B;
    const int tid  = threadIdx.x;
    const int wave = tid >> 5;
    const int lane = tid & 31;
    const int half = lane >> 4;              // 0 -> rows {i,i+1}, 1 -> rows {i+2,i+3}
    const int col  = (wave << 4) + (lane & 15);  // this lane's d-column (B matrix N=lane%16)

    const float* base = x + ((size_t)b * NKP + (size_t)g * ROWS_PER_WG) * DIM + col;

    v2f a; a.x = 1.0f; a.y = 1.0f;   // A = ones(16x4): every lane, both K slots
    v8f c = {};                      // f32 16x16 accumulator (column sums)
    float ssq = 0.0f;

    for (int i = 0; i < ROWS_PER_WG; i += 4) {
        const float* p = base + (size_t)(i + 2 * half) * DIM;
        float x0 = p[0];
        float x1 = p[DIM];
        if (i + 16 < ROWS_PER_WG)
            __builtin_prefetch((const void*)(p + 16 * DIM), 0, 1);  // global_prefetch_b8
        ssq = fmaf(x0, x0, ssq);
        ssq = fmaf(x1, x1, ssq);
        v2f bm; bm.x = x0; bm.y = x1;   // B = 4x16 tile: lane%16 = column, 4 rows across halves/VGPRs
        c = __builtin_amdgcn_wmma_f32_16x16x4_f32(
                /*neg_a=*/false, a, /*neg_b=*/false, bm,
                /*c_mod=*/(short)0, c, /*reuse_a=*/false, /*reuse_b=*/false);
    }

    // D VGPR0: lanes 0..15 hold (M=0, N=lane) = column sum for this wave's columns.
    if (lane < 16)
        ws_s[(size_t)wg * DIM + col] = c[0];

    // Reduce sum(x^2) across the workgroup (fixed order -> deterministic).
    for (int o = 16; o > 0; o >>= 1)
        ssq += __shfl_down(ssq, o, 32);
    __shared__ float red[WAVES];
    if (lane == 0) red[wave] = ssq;
    __syncthreads();
    if (tid == 0) {
        float t = 0.0f;
        for (int wv = 0; wv < WAVES; ++wv) t += red[wv];
        ws_ssq[wg] = t;
    }
}

// Phase 2: single block. Combine the 32 partial column sums per (b,d),
// square, reduce; reduce partial sum(x^2); emit exp(-scale * sep_loss).
__global__ __launch_bounds__(256)
void sep_final_kernel(const float* __restrict__ ws_s,
                      const float* __restrict__ ws_ssq,
                      float* __restrict__ out)
{
    const int tid = threadIdx.x;

    float s2 = 0.0f;
    for (int p = tid; p < BATCH * DIM; p += 256) {       // 4096 (b,d) pairs
        const int b = p >> 7;            // DIM = 128
        const int d = p & (DIM - 1);
        float acc = 0.0f;
        for (int g = 0; g < WG_PER_B; ++g)
            acc += ws_s[((size_t)(b * WG_PER_B + g)) * DIM + d];
        s2 = fmaf(acc, acc, s2);
    }

    float sq = 0.0f;
    for (int q = tid; q < BATCH * WG_PER_B; q += 256)
        sq += ws_ssq[q];

    __shared__ float sh_s2[256];
    __shared__ float sh_sq[256];
    sh_s2[tid] = s2;
    sh_sq[tid] = sq;
    __syncthreads();
    for (int o = 128; o > 0; o >>= 1) {
        if (tid < o) {
            sh_s2[tid] += sh_s2[tid + o];
            sh_sq[tid] += sh_sq[tid + o];
        }
        __syncthreads();
    }

    if (tid == 0) {
        const float sum_sq = sh_sq[0];
        const float sum_s2 = sh_s2[0];
        const float total  = 2.0f * (float)NKP * sum_sq - 2.0f * sum_s2;
        const float loss   = total / ((float)BATCH * (float)NKP * (float)DIM);
        out[0] = expf(-1e-9f * loss);
    }
}

extern "C" void kernel_launch(void* const* d_in, const int* in_sizes, int n_in,
                              void* d_out, int out_size, void* d_ws, size_t ws_size,
                              hipStream_t stream)
{
    (void)in_sizes; (void)n_in; (void)out_size; (void)ws_size;
    const float* x = (const float*)d_in[0];
    float* out = (float*)d_out;

    float* ws_s   = (float*)d_ws;                                  // 1024*128 f32 = 512 KB
    float* ws_ssq = ws_s + (size_t)BATCH * WG_PER_B * DIM;         // 1024 f32   =   4 KB

    sep_partial_kernel<<<dim3(BATCH * WG_PER_B), dim3(THREADS), 0, stream>>>(x, ws_s, ws_ssq);
    sep_final_kernel<<<dim3(1), dim3(256), 0, stream>>>(ws_s, ws_ssq, out);
}